// SKANLinear_59725815218234
// MI455X (gfx1250) — compile-verified
//
#include <hip/hip_runtime.h>
#include <hip/hip_bf16.h>

// ---------------------------------------------------------------------------
// SKANLinear on MI455X (gfx1250, wave32).
//   out[n,o] = sum_k softplus(xe[n,k]*w[o,k]) - ln2*513
// Decomposition (t = z*log2e):
//   softplus(z)-ln2 = max(z,0) + ln2*(log2(1+exp2(-|t|)) - 1)
//   sum_k max(z,0)  = ln2*0.5*(sum_k t + sum_k |t|)   -> two FP32 WMMA GEMMs
//   T = sum_k log2(1+exp2(-|t|))                      -> VALU/TRANS hot loop
//   out = ln2*(0.5*(St+Sat) + T - KPAD)   (zero pads contribute exactly 0)
//
// TRANS reduction: T is accumulated in the *product domain*:
//   log2 prod_{k in group16} (1+e_k)  ==  sum log2(1+e_k)
// 1+e in (1,2] so a 16-element product is <= 2^16 (no overflow, ~16ulp err).
// Inner loop per element: v_mul + v_fmac + v_exp_f32(-|t| via src mods)
// = 2 VALU + 1 TRANS, with one v_log_f32 per 16 elements.
// ---------------------------------------------------------------------------

typedef float v2f __attribute__((ext_vector_type(2)));
typedef float v8f __attribute__((ext_vector_type(8)));

#define N_ROWS 1024
#define K_IN   512
#define K_REAL 513          // IN + bias column
#define KPAD   528          // padded K: even stride, 33 chunks of 16, 132 chunks of 4
#define N_OUT  256

#define LOG2E 1.4426950408889634f
#define LN2   0.6931471805599453f

// ws layout (floats)
#define WS_XS  0                                   // [1024][528]
#define WS_WP  (N_ROWS * KPAD)                     // [256][528]
#define WS_R   (WS_WP + N_OUT * KPAD)              // [1024][256]

// ---------------------------------------------------------------------------
// Kernel 1: repack weight (256 x 513) -> wp (256 x 528), zero-padded tail.
// ---------------------------------------------------------------------------
__global__ void k_weight_prep(const float* __restrict__ w, float* __restrict__ wp) {
    int idx = blockIdx.x * 256 + threadIdx.x;          // 0 .. 256*528-1
    int o = idx / KPAD;
    int k = idx - o * KPAD;
    wp[idx] = (k < K_REAL) ? w[o * K_REAL + k] : 0.0f;
}

// ---------------------------------------------------------------------------
// Kernel 2: layernorm each row of x, scale by log2e, append bias column
// (= log2e) and zero pad to stride 528.  One block (256 thr) per row.
// ---------------------------------------------------------------------------
__global__ void k_layernorm(const float* __restrict__ x,
                            const float* __restrict__ gamma,
                            const float* __restrict__ beta,
                            float* __restrict__ xs) {
    const int row = blockIdx.x;
    const int t   = threadIdx.x;
    __shared__ float ssum[256];
    __shared__ float ssq[256];

    float a = x[row * K_IN + t];
    float b = x[row * K_IN + 256 + t];
    ssum[t] = a + b;
    ssq[t]  = a * a + b * b;
    __syncthreads();
    #pragma unroll
    for (int s = 128; s > 0; s >>= 1) {
        if (t < s) { ssum[t] += ssum[t + s]; ssq[t] += ssq[t + s]; }
        __syncthreads();
    }
    const float inv = 1.0f / (float)K_IN;
    float mu  = ssum[0] * inv;
    float ms  = ssq[0]  * inv;
    float var = ms - mu * mu;
    float rs  = __builtin_amdgcn_rsqf(var + 1e-5f);    // v_rsq_f32

    float* xr = xs + row * KPAD;
    xr[t]       = ((a - mu) * rs * gamma[t]       + beta[t])       * LOG2E;
    xr[t + 256] = ((b - mu) * rs * gamma[t + 256] + beta[t + 256]) * LOG2E;
    if (t < 16) xr[K_IN + t] = (t == 0) ? LOG2E : 0.0f;   // bias col + zero pad
}

// ---------------------------------------------------------------------------
// Kernel 3: dual FP32 WMMA GEMM.  R[n,o] = 0.5*(sum_k t + sum_k |t|)
//          = sum_k max(t,0), t = xs[n,k]*wp[o,k].
// One wave per 16x16 output tile; K accumulated 4 at a time with
// V_WMMA_F32_16X16X4_F32.  Fragment layouts per CDNA5 ISA 7.12.2:
//   A (16x4):  lane L -> M = L%16, K-pair base = 2*(L/16)  -> contiguous b64
//   B (4x16):  lane L -> N = L%16, K-pair base = 2*(L/16)  -> contiguous b64
// All strides even -> 8-byte aligned v2f loads.
// ---------------------------------------------------------------------------
__global__ void k_wmma_relu_gemm(const float* __restrict__ xs,
                                 const float* __restrict__ wp,
                                 float* __restrict__ R) {
    const int lane  = threadIdx.x & 31;
    const int wave  = threadIdx.x >> 5;
    const int gw    = blockIdx.x * 8 + wave;          // 0 .. 1023
    const int n0    = (gw & 63) << 4;                 // 64 n-tiles
    const int o0    = (gw >> 6) << 4;                 // 16 o-tiles

    const int row16 = lane & 15;
    const int khalf = lane >> 4;                      // 0/1 -> K pair {0,1}/{2,3}

    const float* ap = xs + (n0 + row16) * KPAD + (khalf << 1);
    const float* bp = wp + (o0 + row16) * KPAD + (khalf << 1);

    v8f acc  = {};   // sum t
    v8f acc2 = {};   // sum |t|
    #pragma unroll 4
    for (int kc = 0; kc < KPAD / 4; ++kc) {
        v2f a = *(const v2f*)(ap + 4 * kc);
        v2f b = *(const v2f*)(bp + 4 * kc);
        v2f aa = { __builtin_fabsf(a.x), __builtin_fabsf(a.y) };
        v2f bb = { __builtin_fabsf(b.x), __builtin_fabsf(b.y) };
        acc  = __builtin_amdgcn_wmma_f32_16x16x4_f32(false, a,  false, b,
                                                     (short)0, acc,  false, false);
        acc2 = __builtin_amdgcn_wmma_f32_16x16x4_f32(false, aa, false, bb,
                                                     (short)0, acc2, false, false);
    }
    // C/D layout: VGPR v, lane L -> (M = v + 8*(L/16), N = L%16)
    #pragma unroll
    for (int v = 0; v < 8; ++v) {
        int n = n0 + v + (khalf << 3);
        int o = o0 + row16;
        R[n * N_OUT + o] = 0.5f * (acc[v] + acc2[v]);
    }
}

// ---------------------------------------------------------------------------
// Kernel 4: hot loop.  T[n,o] = sum_k log2(1+exp2(-|t|)) via per-chunk
// products, then combine: out = ln2*(R + T - KPAD).
// 256 threads = 16x16 grid of 2x2 register blocks -> 32n x 32o tile.
// k staged in LDS transposed (k-major, stride 34) -> aligned b64 reads,
// conflict-free fills.  Per element: v_mul + v_exp_f32(-|.|) + v_fmac;
// one v_log_f32 + v_add per accumulator per 16-k chunk.
// ---------------------------------------------------------------------------
__global__ void k_softsum(const float* __restrict__ xs,
                          const float* __restrict__ wp,
                          const float* __restrict__ R,
                          float* __restrict__ out) {
    __shared__ __attribute__((aligned(16))) float sx[16 * 34];  // [k][n], 32n + pad
    __shared__ __attribute__((aligned(16))) float sw[16 * 34];  // [k][o], 32o + pad

    const int t  = threadIdx.x;
    const int bo = blockIdx.x & 7;         // 8 o-tiles
    const int bn = blockIdx.x >> 3;        // 32 n-tiles
    const int n0 = bn << 5;
    const int o0 = bo << 5;

    const int j = t & 15;                  // k within chunk (fill phase)
    const int i = t >> 4;                  // row within tile (fill phase)
    const int tx = t & 15;                 // o sub-tile (compute phase)
    const int ty = t >> 4;                 // n sub-tile (compute phase)

    float a00 = 0.f, a01 = 0.f, a10 = 0.f, a11 = 0.f;

    for (int c = 0; c < KPAD / 16; ++c) {
        const int k0 = c * 16;
        // stage xs tile (32n x 16k) and wp tile (32o x 16k), transposed
        sx[j * 34 + i]      = xs[(n0 + i)      * KPAD + k0 + j];
        sx[j * 34 + i + 16] = xs[(n0 + i + 16) * KPAD + k0 + j];
        sw[j * 34 + i]      = wp[(o0 + i)      * KPAD + k0 + j];
        sw[j * 34 + i + 16] = wp[(o0 + i + 16) * KPAD + k0 + j];
        __syncthreads();

        // product-domain accumulation: P = prod (1+e), P <= 2^16
        float p00 = 1.f, p01 = 1.f, p10 = 1.f, p11 = 1.f;
        #pragma unroll
        for (int kk = 0; kk < 16; ++kk) {
            v2f xv = *(const v2f*)&sx[kk * 34 + (ty << 1)];
            v2f wv = *(const v2f*)&sw[kk * 34 + (tx << 1)];
            float t00 = xv.x * wv.x;
            float t01 = xv.x * wv.y;
            float t10 = xv.y * wv.x;
            float t11 = xv.y * wv.y;
            float e00 = __builtin_amdgcn_exp2f(-__builtin_fabsf(t00));
            float e01 = __builtin_amdgcn_exp2f(-__builtin_fabsf(t01));
            float e10 = __builtin_amdgcn_exp2f(-__builtin_fabsf(t10));
            float e11 = __builtin_amdgcn_exp2f(-__builtin_fabsf(t11));
            p00 = __builtin_fmaf(p00, e00, p00);   // p *= (1+e)
            p01 = __builtin_fmaf(p01, e01, p01);
            p10 = __builtin_fmaf(p10, e10, p10);
            p11 = __builtin_fmaf(p11, e11, p11);
        }
        a00 += __builtin_amdgcn_logf(p00);         // v_log_f32 == log2
        a01 += __builtin_amdgcn_logf(p01);
        a10 += __builtin_amdgcn_logf(p10);
        a11 += __builtin_amdgcn_logf(p11);
        __syncthreads();
    }

    const int n = n0 + (ty << 1);
    const int o = o0 + (tx << 1);
    const float bias = -(float)KPAD;
    out[n * N_OUT + o]             = LN2 * (R[n * N_OUT + o]             + a00 + bias);
    out[n * N_OUT + o + 1]         = LN2 * (R[n * N_OUT + o + 1]         + a01 + bias);
    out[(n + 1) * N_OUT + o]       = LN2 * (R[(n + 1) * N_OUT + o]       + a10 + bias);
    out[(n + 1) * N_OUT + o + 1]   = LN2 * (R[(n + 1) * N_OUT + o + 1]   + a11 + bias);
}

// ---------------------------------------------------------------------------
extern "C" void kernel_launch(void* const* d_in, const int* in_sizes, int n_in,
                              void* d_out, int out_size, void* d_ws, size_t ws_size,
                              hipStream_t stream) {
    const float* x     = (const float*)d_in[0];   // 1024 x 512
    const float* w     = (const float*)d_in[1];   // 256 x 513
    const float* gamma = (const float*)d_in[2];   // 512
    const float* beta  = (const float*)d_in[3];   // 512
    float* out = (float*)d_out;                   // 1024 x 256

    float* ws = (float*)d_ws;                     // needs ~3.75 MB
    float* xs = ws + WS_XS;
    float* wp = ws + WS_WP;
    float* R  = ws + WS_R;

    k_weight_prep<<<(N_OUT * KPAD) / 256, 256, 0, stream>>>(w, wp);
    k_layernorm<<<N_ROWS, 256, 0, stream>>>(x, gamma, beta, xs);
    k_wmma_relu_gemm<<<(N_ROWS / 16) * (N_OUT / 16) / 8, 256, 0, stream>>>(xs, wp, R);
    k_softsum<<<(N_ROWS / 32) * (N_OUT / 32), 256, 0, stream>>>(xs, wp, R, out);
}